// FilterbankAttention_54065048322920
// MI455X (gfx1250) — compile-verified
//
#include <hip/hip_runtime.h>

typedef float v2f __attribute__((ext_vector_type(2)));
typedef float v8f __attribute__((ext_vector_type(8)));

#define NF    10
#define ADIM  384
#define BDIM  384
#define HDIM  1024
#define NBATCH 512

#define FY_PITCH 388   // break 384 % 64banks == 0 conflict on A-operand LDS reads
#define GS_PITCH 12    // K padded 10 -> 12 (3 WMMA K-steps of 4)
#define FXW_ROWS 12

// Build one normalized Gaussian filterbank in LDS (rows >= NF are zero pad).
__device__ __forceinline__ void fb_fill(float* F, int pitch, int rows_alloc,
                                        float g, float dstep, float inv2var,
                                        float* s_sum)
{
    const int tid = threadIdx.x;
    if (tid == 0) *s_sum = 0.0f;
    __syncthreads();
    float part = 0.0f;
    const int total = rows_alloc * ADIM;
    for (int idx = tid; idx < total; idx += blockDim.x) {
        const int r = idx / ADIM, c = idx - r * ADIM;
        float v = 0.0f;
        if (r < NF) {
            const float mu = g + ((float)r - 5.5f) * dstep;   // i - N/2 - 0.5
            const float t  = (float)c - mu;
            v = __expf(-t * t * inv2var);
        }
        F[r * pitch + c] = v;
        part += v;
    }
    atomicAdd(s_sum, part);
    __syncthreads();
    const float scale = 1.0f / *s_sum;   // global-sum normalization (as reference)
    for (int idx = tid; idx < total; idx += blockDim.x) {
        const int r = idx / ADIM, c = idx - r * ADIM;
        F[r * pitch + c] *= scale;
    }
    __syncthreads();
}

// ---------------- read path: one block per sample ----------------
__global__ __launch_bounds__(256)
void fba_read_kernel(const float* __restrict__ x, const float* __restrict__ xh,
                     const float* __restrict__ h, const float* __restrict__ Wr,
                     const float* __restrict__ br, float* __restrict__ ro)
{
    __shared__ float FYs[16 * FY_PITCH];   // A operand, rows 10..15 zero
    __shared__ float FXs[NF * ADIM];
    __shared__ float T1s[16 * ADIM];       // F_Y @ x staging
    __shared__ float prs[5];
    __shared__ float ssum;

    const int s    = blockIdx.x;
    const int tid  = threadIdx.x;
    const int lane = tid & 31, wave = tid >> 5;
    const int half = lane >> 4, l16 = lane & 15;

    // ---- pr = h[s] @ W_read + b (5 outputs, K=1024) ----
    if (tid < 5) prs[tid] = br[tid];
    __syncthreads();
    {
        float a0 = 0, a1 = 0, a2 = 0, a3 = 0, a4 = 0;
        const float* hp = h + (size_t)s * HDIM;
        for (int i = tid; i < HDIM; i += 256) {
            const float hv = hp[i];
            const float* w = Wr + (size_t)i * 5;
            a0 += hv * w[0]; a1 += hv * w[1]; a2 += hv * w[2];
            a3 += hv * w[3]; a4 += hv * w[4];
        }
        for (int off = 16; off > 0; off >>= 1) {
            a0 += __shfl_down(a0, off, 32);
            a1 += __shfl_down(a1, off, 32);
            a2 += __shfl_down(a2, off, 32);
            a3 += __shfl_down(a3, off, 32);
            a4 += __shfl_down(a4, off, 32);
        }
        if (lane == 0) {
            atomicAdd(&prs[0], a0); atomicAdd(&prs[1], a1);
            atomicAdd(&prs[2], a2); atomicAdd(&prs[3], a3);
            atomicAdd(&prs[4], a4);
        }
    }
    __syncthreads();

    const float p0      = prs[0], p1 = prs[1];
    const float var     = __expf(prs[2] + 1e-8f);
    const float inv2var = 0.5f / var;
    const float dstep   = __expf(prs[3]) * ((float)(ADIM - 1) / (float)(NF - 1));
    const float gamma   = __expf(prs[4]);
    const float gX = (float)(ADIM + 1) * (p0 + 1.0f) * 0.5f;
    const float gY = (float)(BDIM + 1) * (p1 + 1.0f) * 0.5f;

    fb_fill(FXs, ADIM, NF, gX, dstep, inv2var, &ssum);
    fb_fill(FYs, FY_PITCH, 16, gY, dstep, inv2var, &ssum);

    for (int pass = 0; pass < 2; ++pass) {
        const float* src = (pass == 0 ? x : xh) + (size_t)s * BDIM * ADIM;

        // T1(16x384) = F_Y(16x384, zero-padded) @ x(384x384), WMMA f32 16x16x4
        for (int ct = wave; ct < ADIM / 16; ct += 8) {
            const int c0 = ct * 16;
            v8f acc = {0, 0, 0, 0, 0, 0, 0, 0};
            for (int k0 = 0; k0 < BDIM; k0 += 4) {
                const int kk = k0 + half * 2;
                v2f a, b;
                a.x = FYs[l16 * FY_PITCH + kk];            // A: row=l16, K per half
                a.y = FYs[l16 * FY_PITCH + kk + 1];
                b.x = src[(size_t)kk * ADIM + c0 + l16];   // B: col=l16, K rows
                b.y = src[(size_t)(kk + 1) * ADIM + c0 + l16];
                acc = __builtin_amdgcn_wmma_f32_16x16x4_f32(
                        false, a, false, b, (short)0, acc, false, false);
            }
#pragma unroll
            for (int v = 0; v < 8; ++v)                    // C layout -> LDS
                T1s[(v + half * 8) * ADIM + c0 + l16] = acc[v];
        }
        __syncthreads();

        // T2[n][m] = sum_c T1[n][c] * F_X[m][c]; out = gamma * T2 (10x10, VALU)
        if (tid < NF * NF) {
            const int n = tid / NF, m = tid - n * NF;
            const float* t1 = &T1s[n * ADIM];
            const float* fx = &FXs[m * ADIM];
            float acc2 = 0.0f;
            for (int c = 0; c < ADIM; c += 4) {
                acc2 += t1[c]     * fx[c]     + t1[c + 1] * fx[c + 1]
                      + t1[c + 2] * fx[c + 2] + t1[c + 3] * fx[c + 3];
            }
            ro[(size_t)s * (2 * NF * NF) + pass * (NF * NF) + n * NF + m] =
                gamma * acc2;
        }
        __syncthreads();
    }
}

// ---------------- write path: one block per sample ----------------
__global__ __launch_bounds__(256)
void fba_write_kernel(const float* __restrict__ hdec, const float* __restrict__ Wr,
                      const float* __restrict__ br, const float* __restrict__ Ww,
                      const float* __restrict__ bw, float* __restrict__ wo)
{
    __shared__ float hs[HDIM];
    __shared__ float FYw[NF * BDIM];
    __shared__ float FXw[FXW_ROWS * ADIM];     // rows 10..11 zero (K pad)
    __shared__ float Gs[BDIM * GS_PITCH];      // (1/gamma) * F_Yw^T @ w_t, K pad
    __shared__ float wts[NF * NF];
    __shared__ float pws[5];
    __shared__ float ssum;

    const int s    = blockIdx.x;
    const int tid  = threadIdx.x;
    const int lane = tid & 31, wave = tid >> 5;
    const int half = lane >> 4, l16 = lane & 15;

    for (int i = tid; i < HDIM; i += 256) hs[i] = hdec[(size_t)s * HDIM + i];
    if (tid < 5) pws[tid] = br[tid];
    __syncthreads();

    {   // pw = h_dec @ W_read + b
        float a0 = 0, a1 = 0, a2 = 0, a3 = 0, a4 = 0;
        for (int i = tid; i < HDIM; i += 256) {
            const float hv = hs[i];
            const float* w = Wr + (size_t)i * 5;
            a0 += hv * w[0]; a1 += hv * w[1]; a2 += hv * w[2];
            a3 += hv * w[3]; a4 += hv * w[4];
        }
        for (int off = 16; off > 0; off >>= 1) {
            a0 += __shfl_down(a0, off, 32);
            a1 += __shfl_down(a1, off, 32);
            a2 += __shfl_down(a2, off, 32);
            a3 += __shfl_down(a3, off, 32);
            a4 += __shfl_down(a4, off, 32);
        }
        if (lane == 0) {
            atomicAdd(&pws[0], a0); atomicAdd(&pws[1], a1);
            atomicAdd(&pws[2], a2); atomicAdd(&pws[3], a3);
            atomicAdd(&pws[4], a4);
        }
    }
    __syncthreads();

    const float p0      = pws[0], p1 = pws[1];
    const float var     = __expf(pws[2] + 1e-8f);
    const float inv2var = 0.5f / var;
    const float dstep   = __expf(pws[3]) * ((float)(ADIM - 1) / (float)(NF - 1));
    const float gw_inv  = __expf(-pws[4]);
    const float gX = (float)(ADIM + 1) * (p0 + 1.0f) * 0.5f;
    const float gY = (float)(BDIM + 1) * (p1 + 1.0f) * 0.5f;

    // w_t = h_dec @ W_write + b_write  (100 outputs; W_write stays in L2)
    if (tid < NF * NF) {
        float acc = bw[tid];
        for (int i = 0; i < HDIM; ++i)
            acc += hs[i] * Ww[(size_t)i * (NF * NF) + tid];
        wts[tid] = acc;
    }
    __syncthreads();

    fb_fill(FXw, ADIM, FXW_ROWS, gX, dstep, inv2var, &ssum);
    fb_fill(FYw, BDIM, NF, gY, dstep, inv2var, &ssum);

    // G[j][m] = (1/gamma) * sum_n F_Yw[n][j] * w_t[n][m], K padded to 12
    for (int idx = tid; idx < BDIM * GS_PITCH; idx += 256) {
        const int j = idx / GS_PITCH, m = idx - j * GS_PITCH;
        float v = 0.0f;
        if (m < NF) {
            float acc = 0.0f;
#pragma unroll
            for (int n = 0; n < NF; ++n)
                acc += FYw[n * BDIM + j] * wts[n * NF + m];
            v = acc * gw_inv;
        }
        Gs[idx] = v;
    }
    __syncthreads();

    // write_out(384x384) = G(384x12) @ F_Xw(12x384), WMMA f32 16x16x4
    float* out = wo + (size_t)s * BDIM * ADIM;
    for (int t = wave; t < (BDIM / 16) * (ADIM / 16); t += 8) {
        const int mt = t / (ADIM / 16), nt = t - mt * (ADIM / 16);
        const int j0 = mt * 16, c0 = nt * 16;
        v8f acc = {0, 0, 0, 0, 0, 0, 0, 0};
#pragma unroll
        for (int ks = 0; ks < 12; ks += 4) {
            const int kk = ks + half * 2;
            v2f a, b;
            a.x = Gs[(j0 + l16) * GS_PITCH + kk];
            a.y = Gs[(j0 + l16) * GS_PITCH + kk + 1];
            b.x = FXw[kk * ADIM + c0 + l16];
            b.y = FXw[(kk + 1) * ADIM + c0 + l16];
            acc = __builtin_amdgcn_wmma_f32_16x16x4_f32(
                    false, a, false, b, (short)0, acc, false, false);
        }
#pragma unroll
        for (int v = 0; v < 8; ++v)   // C layout -> coalesced 64B half-wave stores
            out[(size_t)(j0 + v + half * 8) * ADIM + c0 + l16] = acc[v];
    }
}

extern "C" void kernel_launch(void* const* d_in, const int* in_sizes, int n_in,
                              void* d_out, int out_size, void* d_ws, size_t ws_size,
                              hipStream_t stream) {
    (void)in_sizes; (void)n_in; (void)out_size; (void)d_ws; (void)ws_size;
    const float* x    = (const float*)d_in[0];
    const float* xhat = (const float*)d_in[1];
    const float* h    = (const float*)d_in[2];
    const float* hdec = (const float*)d_in[3];
    const float* Wr   = (const float*)d_in[4];
    const float* br   = (const float*)d_in[5];
    const float* Ww   = (const float*)d_in[6];
    const float* bw   = (const float*)d_in[7];

    float* read_out  = (float*)d_out;                       // (512, 20, 10)
    float* write_out = read_out + (size_t)NBATCH * 2 * NF * NF; // (512, 384, 384)

    fba_read_kernel<<<NBATCH, 256, 0, stream>>>(x, xhat, h, Wr, br, read_out);
    fba_write_kernel<<<NBATCH, 256, 0, stream>>>(hdec, Wr, br, Ww, bw, write_out);
}